// AttentionBlockSE3_74406013435999
// MI455X (gfx1250) — compile-verified
//
#include <hip/hip_runtime.h>
#include <hip/hip_bf16.h>
#include <math.h>

typedef float v2f __attribute__((ext_vector_type(2)));
typedef float v4f __attribute__((ext_vector_type(4)));
typedef float v8f __attribute__((ext_vector_type(8)));

#define NUM_HEADS 8
#define HEAD_DIM  16
#define FEATS     128
#define SCALE_INV_SQRT 0.088388347648318447f  /* 1/sqrt(128) */
#define TILE_E    16
#define ROW_F     132      /* padded LDS row stride (floats): 16B-aligned, conflict-free */
#define WAVES_A   2

// ---- float atomic helpers -------------------------------------------------
__device__ __forceinline__ void atomicMaxF32(float* addr, float v) {
  int iv = __float_as_int(v);
  if (iv >= 0) atomicMax((int*)addr, iv);                 // -> global_atomic_max_i32
  else         atomicMin((unsigned int*)addr, (unsigned int)iv); // -> global_atomic_min_u32
}
// Non-returning native f32 add (TH[0]=0): STOREcnt-tracked, no VGPR writeback,
// no CAS-loop risk. Same operand format the compiler emits for min/max above.
__device__ __forceinline__ void atomicAddF32(float* addr, float v) {
  asm volatile("global_atomic_add_f32 %0, %1, off scope:SCOPE_DEV"
               :: "v"(addr), "v"(v) : "memory");
}

// ---- pass 0: init out/smax/ssum, build contiguous qcat[N,128] -------------
__global__ void k_init(const float* __restrict__ q0, const float* __restrict__ q1,
                       float* __restrict__ qcat, float* __restrict__ smax,
                       float* __restrict__ ssum, float* __restrict__ out, int N) {
  int i = blockIdx.x * blockDim.x + threadIdx.x;
  if (i < N * FEATS) out[i] = 0.0f;
  if (i < N * 8) { smax[i] = -INFINITY; ssum[i] = 0.0f; }
  if (i < N * 32) {
    int n = i >> 5, ch = i & 31;
    v4f v;
    v.x = q0[i];
    v.y = q1[i * 3 + 0];
    v.z = q1[i * 3 + 1];
    v.w = q1[i * 3 + 2];
    *(v4f*)(qcat + (size_t)n * FEATS + ch * 4) = v;
  }
}

// ---- pass 1: scores[e,h] = (k[e,h,:] . q[dst[e],h,:]) / sqrt(128); smax ----
__global__ void k_scores(const float* __restrict__ key,
                         const float* __restrict__ qcat,
                         const long long* __restrict__ dst,
                         float* __restrict__ scores,
                         float* __restrict__ smax,
                         long long E) {
  __shared__ float lds[WAVES_A][2][TILE_E * ROW_F];
  const int lane = threadIdx.x & 31;
  const int wv   = threadIdx.x >> 5;
  const long long tile = (long long)blockIdx.x * WAVES_A + wv;
  const long long e0 = tile * TILE_E;
  if (e0 >= E) return;

  const int m = lane & 15;          // edge-in-tile (row M / col N)
  const int p = lane >> 4;          // feature-pair half
  const long long myE = e0 + m;
  const int myDst = (myE < E) ? (int)dst[myE] : 0;

  if (e0 + TILE_E <= E) {
    float* kT = &lds[wv][0][0];
    float* qT = &lds[wv][1][0];
    // --- async stage K tile (streaming: non-temporal) ---
#pragma unroll
    for (int r = 0; r < TILE_E; ++r) {
      unsigned ldsOff = (unsigned)(uintptr_t)&kT[r * ROW_F] + (unsigned)lane * 16u;
      const char* g = (const char*)key + ((size_t)(e0 + r) * FEATS) * 4 + lane * 16;
      asm volatile("global_load_async_to_lds_b128 %0, %1, off th:TH_LOAD_NT"
                   :: "v"(ldsOff), "v"(g) : "memory");
    }
    // --- async stage gathered Q rows (L2-resident) ---
#pragma unroll
    for (int r = 0; r < TILE_E; ++r) {
      int dr = __builtin_amdgcn_readlane(myDst, r);
      unsigned ldsOff = (unsigned)(uintptr_t)&qT[r * ROW_F] + (unsigned)lane * 16u;
      const char* g = (const char*)qcat + (size_t)dr * FEATS * 4 + lane * 16;
      asm volatile("global_load_async_to_lds_b128 %0, %1, off"
                   :: "v"(ldsOff), "v"(g) : "memory");
    }
    asm volatile("s_wait_asynccnt 0x0" ::: "memory");

#pragma unroll
    for (int h = 0; h < NUM_HEADS; ++h) {
      v8f acc = {0.f, 0.f, 0.f, 0.f, 0.f, 0.f, 0.f, 0.f};
#pragma unroll
      for (int c = 0; c < 4; ++c) {
        const int f = h * HEAD_DIM + c * 4 + 2 * p;
        v2f a, b;
        a.x = kT[m * ROW_F + f];  a.y = kT[m * ROW_F + f + 1];
        b.x = qT[m * ROW_F + f];  b.y = qT[m * ROW_F + f + 1];
        acc = __builtin_amdgcn_wmma_f32_16x16x4_f32(
                  false, a, false, b, (short)0, acc, false, false);
      }
      // diag(D): i<8 -> (lane i, vgpr i); i>=8 -> (lane i+16, vgpr i-8)
      const int rs = lane & 7;
      float dv = acc[0];
      dv = (rs == 1) ? acc[1] : dv;
      dv = (rs == 2) ? acc[2] : dv;
      dv = (rs == 3) ? acc[3] : dv;
      dv = (rs == 4) ? acc[4] : dv;
      dv = (rs == 5) ? acc[5] : dv;
      dv = (rs == 6) ? acc[6] : dv;
      dv = (rs == 7) ? acc[7] : dv;
      if (lane < 8 || lane >= 24) {            // these lanes own diag of edge m
        float s = dv * SCALE_INV_SQRT;
        scores[myE * 8 + h] = s;
        atomicMaxF32(&smax[(size_t)myDst * 8 + h], s);
      }
    }
  } else {
    // scalar tail (no WMMA: EXEC may be partial here)
    if (p == 0 && myE < E) {
      for (int h = 0; h < NUM_HEADS; ++h) {
        float s = 0.f;
        for (int j = 0; j < HEAD_DIM; ++j)
          s += key[(size_t)myE * FEATS + h * HEAD_DIM + j] *
               qcat[(size_t)myDst * FEATS + h * HEAD_DIM + j];
        s *= SCALE_INV_SQRT;
        scores[myE * 8 + h] = s;
        atomicMaxF32(&smax[(size_t)myDst * 8 + h], s);
      }
    }
  }
}

// ---- pass 2: ex = exp(s - smax[dst]); ssum += ex (in-place into scores) ----
__global__ void k_softmax(const long long* __restrict__ dst,
                          float* __restrict__ scores,
                          const float* __restrict__ smax,
                          float* __restrict__ ssum,
                          long long E) {
  long long i = (long long)blockIdx.x * blockDim.x + threadIdx.x;
  if (i >= E * 8) return;
  long long e = i >> 3;
  int h = (int)(i & 7);
  int d = (int)dst[e];
  float ex = expf(scores[i] - smax[(size_t)d * 8 + h]);
  scores[i] = ex;
  atomicAddF32(&ssum[(size_t)d * 8 + h], ex);
}

// ---- pass 3: out[dst] += (ex/ssum[dst]) * v   (wave per edge, lane=channel)
__global__ void k_aggregate(const float* __restrict__ value,
                            const long long* __restrict__ dst,
                            const float* __restrict__ ex,
                            const float* __restrict__ ssum,
                            float* __restrict__ out,
                            long long E) {
  const int lane = threadIdx.x & 31;                 // channel 0..31
  long long e = (long long)blockIdx.x * (blockDim.x >> 5) + (threadIdx.x >> 5);
  if (e >= E) return;
  int d = (int)dst[e];
  int h = lane >> 2;
  float w = ex[e * 8 + h] / ssum[(size_t)d * 8 + h];
  v4f v = __builtin_nontemporal_load((const v4f*)(value + (size_t)e * FEATS) + lane);
  float* o = out + (size_t)d * FEATS + lane * 4;
  atomicAddF32(o + 0, w * v.x);
  atomicAddF32(o + 1, w * v.y);
  atomicAddF32(o + 2, w * v.z);
  atomicAddF32(o + 3, w * v.w);
}

extern "C" void kernel_launch(void* const* d_in, const int* in_sizes, int n_in,
                              void* d_out, int out_size, void* d_ws, size_t ws_size,
                              hipStream_t stream) {
  const float* value = (const float*)d_in[0];
  const float* key   = (const float*)d_in[1];
  const float* q0    = (const float*)d_in[2];
  const float* q1    = (const float*)d_in[3];
  const long long* ei = (const long long*)d_in[4];   // int64 per reference

  const long long E = in_sizes[0] / FEATS;           // value: E*32*4
  const int       N = in_sizes[2] / 32;              // query_0: N*32*1
  const long long* dst = ei + E;                     // edge_index[1]

  char* ws = (char*)d_ws;
  float* scores = (float*)ws;                                        // E*8
  float* qcat   = (float*)(ws + (size_t)E * 8 * 4);                  // N*128
  float* smax   = (float*)(ws + (size_t)E * 8 * 4 + (size_t)N * FEATS * 4); // N*8
  float* ssum   = smax + (size_t)N * 8;                              // N*8
  float* out    = (float*)d_out;

  {
    int total = N * FEATS, bs = 256;
    k_init<<<(total + bs - 1) / bs, bs, 0, stream>>>(q0, q1, qcat, smax, ssum, out, N);
  }
  {
    long long tiles = (E + TILE_E - 1) / TILE_E;
    int gs = (int)((tiles + WAVES_A - 1) / WAVES_A);
    k_scores<<<gs, 32 * WAVES_A, 0, stream>>>(key, qcat, dst, scores, smax, E);
  }
  {
    long long tot = E * 8; int bs = 256;
    k_softmax<<<(int)((tot + bs - 1) / bs), bs, 0, stream>>>(dst, scores, smax, ssum, E);
  }
  {
    int bs = 256;                                     // 8 edges / block
    k_aggregate<<<(int)((E + 7) / 8), bs, 0, stream>>>(value, dst, scores, ssum, out, E);
  }
}